// GAT_59373627900055
// MI455X (gfx1250) — compile-verified
//
#include <hip/hip_runtime.h>
#include <hip/hip_bf16.h>

#define N_NODES 100000
#define N_EDGES 1600000
#define EP      (N_EDGES + N_NODES)   // edges + self loops = 1,700,000
#define F_IN    128
#define HID     16
#define HEADS   4
#define C1      (HEADS * HID)         // 64
#define NCLS    40
#define NEG_SLOPE 0.2f

typedef float v2f __attribute__((ext_vector_type(2)));
typedef float v8f __attribute__((ext_vector_type(8)));

// ---------------------------------------------------------------------------
// GEMM1: H[N,64] = X[N,128] @ W1[64,128]^T   (fp32 WMMA 16x16x4, exact f32)
// One wave per 16-row STRIP: the A fragment is loaded once per K-step and
// reused across all 4 column tiles (4 accumulators), so X is read exactly
// once from HBM. W1 (32 KB) stays L2-resident.
// A frag (16x4 f32): lanes 0-15 -> K={0,1}, lanes 16-31 -> K={2,3}, M = lane%16
// B frag (4x16 f32): lanes 0-15 -> K={0,1}, lanes 16-31 -> K={2,3}, N = lane%16
// C frag: VGPR r -> M = r + 8*(lane/16), N = lane%16
// ---------------------------------------------------------------------------
__global__ __launch_bounds__(256) void gemm1_wmma(const float* __restrict__ X,
                                                  const float* __restrict__ W,
                                                  float* __restrict__ H) {
  const int wave = (blockIdx.x * 256 + (int)threadIdx.x) >> 5;
  if (wave >= N_NODES / 16) return;         // uniform per wave
  const int lane = threadIdx.x & 31;
  const int l15 = lane & 15;
  const int khalf = (lane >> 4) << 1;       // 0 or 2
  const int row0 = wave * 16;
  const float* __restrict__ xrow = X + (size_t)(row0 + l15) * F_IN;
  const float* __restrict__ wrow = W + (size_t)l15 * F_IN;
  v8f acc0 = {}, acc1 = {}, acc2 = {}, acc3 = {};
#pragma unroll
  for (int k = 0; k < F_IN; k += 4) {
    v2f a;
    a.x = xrow[k + khalf];  a.y = xrow[k + khalf + 1];
    v2f b0, b1, b2, b3;
    b0.x = wrow[k + khalf];                b0.y = wrow[k + khalf + 1];
    b1.x = wrow[16 * F_IN + k + khalf];    b1.y = wrow[16 * F_IN + k + khalf + 1];
    b2.x = wrow[32 * F_IN + k + khalf];    b2.y = wrow[32 * F_IN + k + khalf + 1];
    b3.x = wrow[48 * F_IN + k + khalf];    b3.y = wrow[48 * F_IN + k + khalf + 1];
    acc0 = __builtin_amdgcn_wmma_f32_16x16x4_f32(false, a, false, b0, (short)0,
                                                 acc0, false, false);
    acc1 = __builtin_amdgcn_wmma_f32_16x16x4_f32(false, a, false, b1, (short)0,
                                                 acc1, false, false);
    acc2 = __builtin_amdgcn_wmma_f32_16x16x4_f32(false, a, false, b2, (short)0,
                                                 acc2, false, false);
    acc3 = __builtin_amdgcn_wmma_f32_16x16x4_f32(false, a, false, b3, (short)0,
                                                 acc3, false, false);
  }
  const int mbase = row0 + ((lane >> 4) << 3);
#pragma unroll
  for (int r = 0; r < 8; ++r) {
    float* hr = H + (size_t)(mbase + r) * C1 + l15;
    hr[0]  = acc0[r];
    hr[16] = acc1[r];
    hr[32] = acc2[r];
    hr[48] = acc3[r];
  }
}

// ---------------------------------------------------------------------------
// GEMM2: H[N,40] = X[N,64] @ W2[40,64]^T   (fp32 WMMA, 3 col tiles per wave,
// branchless clamp keeps EXEC all-ones through the WMMAs; cols >= 40 dropped)
// ---------------------------------------------------------------------------
__global__ __launch_bounds__(256) void gemm2_wmma(const float* __restrict__ X,
                                                  const float* __restrict__ W,
                                                  float* __restrict__ H) {
  const int wave = (blockIdx.x * 256 + (int)threadIdx.x) >> 5;
  if (wave >= N_NODES / 16) return;         // uniform per wave
  const int lane = threadIdx.x & 31;
  const int l15 = lane & 15;
  const int khalf = (lane >> 4) << 1;
  const int row0 = wave * 16;
  const float* __restrict__ xrow = X + (size_t)(row0 + l15) * C1;
  const int wc2 = (32 + l15 > NCLS - 1) ? (NCLS - 1) : (32 + l15);
  const float* __restrict__ w0 = W + (size_t)l15 * C1;
  const float* __restrict__ w1 = W + (size_t)(16 + l15) * C1;
  const float* __restrict__ w2 = W + (size_t)wc2 * C1;
  v8f acc0 = {}, acc1 = {}, acc2 = {};
#pragma unroll
  for (int k = 0; k < C1; k += 4) {
    v2f a;
    a.x = xrow[k + khalf];  a.y = xrow[k + khalf + 1];
    v2f b0, b1, b2;
    b0.x = w0[k + khalf];  b0.y = w0[k + khalf + 1];
    b1.x = w1[k + khalf];  b1.y = w1[k + khalf + 1];
    b2.x = w2[k + khalf];  b2.y = w2[k + khalf + 1];
    acc0 = __builtin_amdgcn_wmma_f32_16x16x4_f32(false, a, false, b0, (short)0,
                                                 acc0, false, false);
    acc1 = __builtin_amdgcn_wmma_f32_16x16x4_f32(false, a, false, b1, (short)0,
                                                 acc1, false, false);
    acc2 = __builtin_amdgcn_wmma_f32_16x16x4_f32(false, a, false, b2, (short)0,
                                                 acc2, false, false);
  }
  const int mbase = row0 + ((lane >> 4) << 3);
#pragma unroll
  for (int r = 0; r < 8; ++r) {
    float* hr = H + (size_t)(mbase + r) * NCLS + l15;
    hr[0]  = acc0[r];
    hr[16] = acc1[r];
  }
  if (32 + l15 < NCLS) {                    // divergence only after all WMMAs
#pragma unroll
    for (int r = 0; r < 8; ++r)
      H[(size_t)(mbase + r) * NCLS + 32 + l15] = acc2[r];
  }
}

// ---- per-node attention logits: alpha_s/d[n,h] = sum_c h[n,h,c]*a[h,c] ----
__global__ __launch_bounds__(256) void alpha_kernel(
    const float* __restrict__ H, const float* __restrict__ a_s,
    const float* __restrict__ a_d, float* __restrict__ os,
    float* __restrict__ od, int nh, int ch) {
  const int n = blockIdx.x * blockDim.x + threadIdx.x;
  if (n >= N_NODES) return;
  const float* hrow = H + (size_t)n * nh * ch;
  for (int h = 0; h < nh; ++h) {
    float ss = 0.f, sd = 0.f;
    for (int c = 0; c < ch; ++c) {
      const float v = hrow[h * ch + c];
      ss += v * a_s[h * ch + c];
      sd += v * a_d[h * ch + c];
    }
    os[n * nh + h] = ss;
    od[n * nh + h] = sd;
  }
}

// order-preserving float<->u32 key so segment-max uses native u32 atomic max
__device__ __forceinline__ unsigned fkey(float f) {
  const unsigned b = __float_as_uint(f);
  return (b & 0x80000000u) ? ~b : (b | 0x80000000u);
}
__device__ __forceinline__ float funkey(unsigned k) {
  return __uint_as_float((k & 0x80000000u) ? (k & 0x7FFFFFFFu) : ~k);
}

__device__ __forceinline__ void edge_ends(const int* __restrict__ ei, int e,
                                          int& s, int& d) {
  if (e < N_EDGES) { s = ei[e]; d = ei[N_EDGES + e]; }
  else             { s = e - N_EDGES; d = s; }      // self loop
}

// ---- pass A: per-dst running max of leaky_relu(as[src]+ad[dst]) ----------
template <int NH>
__global__ __launch_bounds__(256) void edge_max_kernel(
    const int* __restrict__ ei, const float* __restrict__ as,
    const float* __restrict__ ad, unsigned* __restrict__ emax) {
  const int t = blockIdx.x * blockDim.x + threadIdx.x;
  if (t >= EP * NH) return;
  const int e = t / NH, h = t % NH;
  int s, d; edge_ends(ei, e, s, d);
  float v = as[s * NH + h] + ad[d * NH + h];
  v = fmaxf(v, NEG_SLOPE * v);              // leaky_relu (slope < 1)
  atomicMax(&emax[d * NH + h], fkey(v));
}

// ---- pass B: numerators + per-dst denominator -----------------------------
template <int NH>
__global__ __launch_bounds__(256) void edge_exp_kernel(
    const int* __restrict__ ei, const float* __restrict__ as,
    const float* __restrict__ ad, const unsigned* __restrict__ emax,
    float* __restrict__ exv, float* __restrict__ den) {
  const int t = blockIdx.x * blockDim.x + threadIdx.x;
  if (t >= EP * NH) return;
  const int e = t / NH, h = t % NH;
  int s, d; edge_ends(ei, e, s, d);
  float v = as[s * NH + h] + ad[d * NH + h];
  v = fmaxf(v, NEG_SLOPE * v);
  const float ex = expf(v - funkey(emax[d * NH + h]));
  exv[t] = ex;
  atomicAdd(&den[d * NH + h], ex);
}

// ---- pass C: out[dst,c] += h[src,c] * coef(e, head(c)) --------------------
template <int NH, int CH>
__global__ __launch_bounds__(256) void edge_agg_kernel(
    const int* __restrict__ ei, const float* __restrict__ H,
    const float* __restrict__ exv, const float* __restrict__ den,
    float* __restrict__ out) {
  constexpr int CT = NH * CH;
  const long long t = (long long)blockIdx.x * blockDim.x + threadIdx.x;
  if (t >= (long long)EP * CT) return;
  const int e = (int)(t / CT), c = (int)(t % CT);
  const int h = c / CH;
  int s, d; edge_ends(ei, e, s, d);
  const float coef = exv[e * NH + h] / den[d * NH + h];
  atomicAdd(&out[(size_t)d * CT + c], H[(size_t)s * CT + c] * coef);
}

__global__ __launch_bounds__(256) void bias_relu_kernel(
    float* __restrict__ out, const float* __restrict__ b) {
  const int i = blockIdx.x * blockDim.x + threadIdx.x;
  if (i >= N_NODES * C1) return;
  out[i] = fmaxf(out[i] + b[i & (C1 - 1)], 0.f);
}

// ---- final: out += b2, then in-place row log_softmax ----------------------
__global__ __launch_bounds__(256) void logsoftmax_kernel(
    float* __restrict__ out, const float* __restrict__ b2) {
  const int n = blockIdx.x * blockDim.x + threadIdx.x;
  if (n >= N_NODES) return;
  float* row = out + (size_t)n * NCLS;
  float v[NCLS];
  float m = -3.4e38f;
  for (int c = 0; c < NCLS; ++c) { v[c] = row[c] + b2[c]; m = fmaxf(m, v[c]); }
  float ssum = 0.f;
  for (int c = 0; c < NCLS; ++c) ssum += expf(v[c] - m);
  const float ls = logf(ssum) + m;
  for (int c = 0; c < NCLS; ++c) row[c] = v[c] - ls;
}

extern "C" void kernel_launch(void* const* d_in, const int* in_sizes, int n_in,
                              void* d_out, int out_size, void* d_ws,
                              size_t ws_size, hipStream_t stream) {
  const float* x   = (const float*)d_in[0];
  const int*   ei  = (const int*)d_in[1];
  const float* W1  = (const float*)d_in[2];
  const float* as1 = (const float*)d_in[3];
  const float* ad1 = (const float*)d_in[4];
  const float* b1  = (const float*)d_in[5];
  const float* W2  = (const float*)d_in[6];
  const float* as2 = (const float*)d_in[7];
  const float* ad2 = (const float*)d_in[8];
  const float* b2  = (const float*)d_in[9];
  float* out = (float*)d_out;

  char* ws = (char*)d_ws;
  size_t off = 0;
  auto alloc = [&](size_t bytes) -> void* {
    void* p = ws + off;
    off = (off + bytes + 255) & ~(size_t)255;
    return p;
  };
  float*    h1    = (float*)alloc((size_t)N_NODES * C1 * 4);
  float*    agg1  = (float*)alloc((size_t)N_NODES * C1 * 4);
  float*    als1  = (float*)alloc((size_t)N_NODES * HEADS * 4);
  float*    ald1  = (float*)alloc((size_t)N_NODES * HEADS * 4);
  unsigned* emax1 = (unsigned*)alloc((size_t)N_NODES * HEADS * 4);
  float*    den1  = (float*)alloc((size_t)N_NODES * HEADS * 4);
  float*    ex1   = (float*)alloc((size_t)EP * HEADS * 4);
  float*    h2    = (float*)alloc((size_t)N_NODES * NCLS * 4);
  float*    als2  = (float*)alloc((size_t)N_NODES * 4);
  float*    ald2  = (float*)alloc((size_t)N_NODES * 4);
  unsigned* emax2 = (unsigned*)alloc((size_t)N_NODES * 4);
  float*    den2  = (float*)alloc((size_t)N_NODES * 4);
  float*    ex2   = (float*)alloc((size_t)EP * 4);

  // init accumulators (key 0 == most-negative float for the max buffers;
  // every dst has a self loop, so init values are always superseded)
  hipMemsetAsync(agg1,  0, (size_t)N_NODES * C1 * 4, stream);
  hipMemsetAsync(emax1, 0, (size_t)N_NODES * HEADS * 4, stream);
  hipMemsetAsync(den1,  0, (size_t)N_NODES * HEADS * 4, stream);
  hipMemsetAsync(emax2, 0, (size_t)N_NODES * 4, stream);
  hipMemsetAsync(den2,  0, (size_t)N_NODES * 4, stream);
  hipMemsetAsync(out,   0, (size_t)N_NODES * NCLS * 4, stream);

  // ---- layer 1 ----
  {
    const int waves = N_NODES / 16;                     // 6250 strip-waves
    gemm1_wmma<<<(waves + 7) / 8, 256, 0, stream>>>(x, W1, h1);
  }
  alpha_kernel<<<(N_NODES + 255) / 256, 256, 0, stream>>>(h1, as1, ad1, als1,
                                                          ald1, HEADS, HID);
  edge_max_kernel<HEADS><<<(EP * HEADS + 255) / 256, 256, 0, stream>>>(
      ei, als1, ald1, emax1);
  edge_exp_kernel<HEADS><<<(EP * HEADS + 255) / 256, 256, 0, stream>>>(
      ei, als1, ald1, emax1, ex1, den1);
  {
    const long long tot = (long long)EP * C1;
    edge_agg_kernel<HEADS, HID><<<(unsigned)((tot + 255) / 256), 256, 0,
                                  stream>>>(ei, h1, ex1, den1, agg1);
  }
  bias_relu_kernel<<<(N_NODES * C1 + 255) / 256, 256, 0, stream>>>(agg1, b1);

  // ---- layer 2 ----
  {
    const int waves = N_NODES / 16;                     // 6250 strip-waves
    gemm2_wmma<<<(waves + 7) / 8, 256, 0, stream>>>(agg1, W2, h2);
  }
  alpha_kernel<<<(N_NODES + 255) / 256, 256, 0, stream>>>(h2, as2, ad2, als2,
                                                          ald2, 1, NCLS);
  edge_max_kernel<1><<<(EP + 255) / 256, 256, 0, stream>>>(ei, als2, ald2,
                                                           emax2);
  edge_exp_kernel<1><<<(EP + 255) / 256, 256, 0, stream>>>(ei, als2, ald2,
                                                           emax2, ex2, den2);
  {
    const long long tot = (long long)EP * NCLS;
    edge_agg_kernel<1, NCLS><<<(unsigned)((tot + 255) / 256), 256, 0, stream>>>(
        ei, h2, ex2, den2, out);
  }
  logsoftmax_kernel<<<(N_NODES + 255) / 256, 256, 0, stream>>>(out, b2);
  (void)in_sizes; (void)n_in; (void)out_size; (void)ws_size;
}